// CensoredRW_91207925498576
// MI455X (gfx1250) — compile-verified
//
#include <hip/hip_runtime.h>

typedef __attribute__((ext_vector_type(2))) float v2f;
typedef __attribute__((ext_vector_type(8))) float v8f;

#define NN   256   // N
#define NB   32    // batches
#define NSEQ 16    // seq_len (n = 15)

// One wave32 per batch.
// Phase 1: rowsum_i = sum_j exp(P[perm[i], perm[j]]) for i=0..15 via
//          V_WMMA_F32_16X16X4_F32 with an all-ones B matrix (64 K-chunks).
// Phase 2: build 15x16 augmented A[i][j] = (i==j) - t[i][j] in LDS,
//          unpivoted Gaussian elimination (M = I - T is an M-matrix: stable),
//          step_k = -U[k][k+1]/U[k][k], lik_b = sum_k log(step_k).
//
// P ~ U(0,1) so exp args are in (0,1): native v_exp (__expf) is exact enough
// and avoids the ~10-op range-checked libm expansion in the hot loop.
__global__ __launch_bounds__(32)
void crw_batch_kernel(const float* __restrict__ P,
                      const int* __restrict__ perm,
                      float* __restrict__ lik /* [NB] */) {
  const int b    = blockIdx.x;
  const int lane = threadIdx.x;   // 0..31, wave32

  __shared__ int   pidx[NN];
  __shared__ float rowsum[16];
  __shared__ float Am[15][17];    // 15x16 augmented matrix (+1 pad)

  // ---- load this batch's permutation row into LDS ----
  const int* pb = perm + b * NN;
#pragma unroll
  for (int c = 0; c < NN / 32; ++c) pidx[lane + 32 * c] = pb[lane + 32 * c];
  __syncthreads();

  // ---- Phase 1: row sums via f32 WMMA against all-ones B ----
  // ISA 16x4 f32 A layout: lanes 0-15 hold (M=lane, K=0/1) in VGPR0/1,
  //                        lanes 16-31 hold (M=lane-16, K=2/3).
  const int arow  = lane & 15;
  const int kbase = (lane < 16) ? 0 : 2;
  const float* Prow = P + (size_t)pidx[arow] * NN;

  v8f acc = {};
  const v2f ones = {1.0f, 1.0f};
  for (int c = 0; c < NN / 4; ++c) {
    const int j0 = 4 * c + kbase;              // even -> int2 is 8B aligned
    const int2 jj = *(const int2*)&pidx[j0];   // single ds_load_b64
    v2f a;
    a.x = __expf(Prow[jj.x]);
    a.y = __expf(Prow[jj.y]);
    // D = A(16x4) * ones(4x16) + C  ->  every column of C holds the row sums
    acc = __builtin_amdgcn_wmma_f32_16x16x4_f32(
        /*neg_a=*/false, a, /*neg_b=*/false, ones,
        /*c_mod=*/(short)0, acc, /*reuse_a=*/false, /*reuse_b=*/false);
  }

  // C layout: VGPR r -> M=r (lanes 0-15) / M=8+r (lanes 16-31); all N equal.
  if (lane == 0) {
#pragma unroll
    for (int r = 0; r < 8; ++r) rowsum[r] = acc[r];
  } else if (lane == 16) {
#pragma unroll
    for (int r = 0; r < 8; ++r) rowsum[8 + r] = acc[r];
  }
  __syncthreads();

  // ---- Phase 2a: build augmented matrix A[i][j] = delta_ij - t[i][j] ----
  // t[i][j] = exp(P[perm[i],perm[j]]) / max(rowsum_i, 1e-12), t[i][i] = 0.
  for (int w = lane; w < 15 * 16; w += 32) {
    const int i = w >> 4;
    const int j = w & 15;
    float t = 0.0f;
    if (i != j)
      t = __expf(P[(size_t)pidx[i] * NN + pidx[j]]) / fmaxf(rowsum[i], 1e-12f);
    Am[i][j] = ((i == j) ? 1.0f : 0.0f) - t;
  }
  __syncthreads();

  // ---- Phase 2b: unpivoted Gaussian elimination on 15x16 ----
  for (int k = 0; k < 14; ++k) {
    const float pivinv = 1.0f / Am[k][k];      // fixed by previous steps
    const int rows = 14 - k, cols = 15 - k;    // i in (k,14], j in (k,15]
    for (int w = lane; w < rows * cols; w += 32) {
      const int i = k + 1 + w / cols;
      const int j = k + 1 + w % cols;
      // reads touch only row k / column k (not written this step): race-free
      Am[i][j] -= Am[i][k] * pivinv * Am[k][j];
    }
    __syncthreads();
  }

  // ---- step_k = -U[k][k+1]/U[k][k];  lik_b = sum_k log(step_k) ----
  float v = 0.0f;
  if (lane < NSEQ - 1) {
    const float stepk = -Am[lane][lane + 1] / Am[lane][lane];
    v = logf(stepk);
  }
#pragma unroll
  for (int off = 16; off > 0; off >>= 1) v += __shfl_down(v, off);
  if (lane == 0) lik[b] = v;
}

// Deterministic final reduction (no float atomics): out = -sum_b lik[b]
__global__ __launch_bounds__(32)
void crw_reduce_kernel(const float* __restrict__ lik, float* __restrict__ out) {
  float v = lik[threadIdx.x];
#pragma unroll
  for (int off = 16; off > 0; off >>= 1) v += __shfl_down(v, off);
  if (threadIdx.x == 0) out[0] = -v;
}

extern "C" void kernel_launch(void* const* d_in, const int* in_sizes, int n_in,
                              void* d_out, int out_size, void* d_ws, size_t ws_size,
                              hipStream_t stream) {
  const float* P    = (const float*)d_in[0];   // [256*256] f32
  const int*   perm = (const int*)d_in[1];     // [32*256] i32
  (void)in_sizes; (void)n_in; (void)out_size; (void)ws_size;

  float* lik = (float*)d_ws;                   // 32 floats of scratch
  crw_batch_kernel<<<NB, 32, 0, stream>>>(P, perm, lik);
  crw_reduce_kernel<<<1, 32, 0, stream>>>(lik, (float*)d_out);
}